// MinibatchDiscrimination_4131758539431
// MI455X (gfx1250) — compile-verified
//
#include <hip/hip_runtime.h>

typedef __attribute__((ext_vector_type(2))) float v2f;
typedef __attribute__((ext_vector_type(8))) float v8f;

#define N_SAMP 256
#define IN_F   1024
#define OUT_F  128
#define KDIM   32
#define COLS   (OUT_F * KDIM)      // 4096
#define OUT_STRIDE (IN_F + OUT_F)  // 1152

// XOR swizzle: row-major 16x1024 tile, column index xored with (row<<2).
// Bits [1:0] of the column are preserved, so float2 (8B) accesses stay
// contiguous, and the 16 lanes of each half-wave hit 16 distinct banks.
#define SWZ(r, c) ((r) * IN_F + ((c) ^ ((r) << 2)))

// ---------------------------------------------------------------------------
// Phase 1: M[n, o*32+k] = sum_i x[n,i] * T[i, o*32+k]
// x:[256,1024]  T viewed as [1024,4096]  M:[256,4096]
// One wave per 16x16 output tile, V_WMMA_F32_16X16X4_F32 over K=1024.
// ---------------------------------------------------------------------------
__global__ __launch_bounds__(128) void mbd_gemm(const float* __restrict__ x,
                                                const float* __restrict__ T,
                                                float* __restrict__ M)
{
    __shared__ float ldsX[16 * IN_F];          // 64 KB x-tile
    const int tid  = threadIdx.x;
    const int wave = tid >> 5;
    const int lane = tid & 31;
    const int rowBase = blockIdx.y * 16;                 // 0..240
    const int colBase = (blockIdx.x * 4 + wave) * 16;    // 0..4080

    // Cooperative, coalesced load of the 16x1024 x-tile (swizzled store).
    for (int idx = tid; idx < 16 * IN_F; idx += 128) {
        const int rr = idx >> 10;
        const int cc = idx & (IN_F - 1);
        ldsX[SWZ(rr, cc)] = x[(rowBase + rr) * IN_F + cc];
    }
    __syncthreads();

    // A-matrix 16x4 f32 layout: lanes 0-15 -> M=r, VGPR0=K0,VGPR1=K1;
    // lanes 16-31 -> M=r, VGPR0=K2,VGPR1=K3.  B mirrors with N across lanes.
    const int r  = lane & 15;
    const int kk = (lane >> 4) << 1;   // 0 or 2

    v8f acc = {0.f, 0.f, 0.f, 0.f, 0.f, 0.f, 0.f, 0.f};

    for (int k0 = 0; k0 < IN_F; k0 += 4) {
        const int kbase = k0 + kk;                 // 0 or 2 mod 4
        // A pair from LDS: swizzle keeps the pair contiguous & 8B aligned.
        const float2 ap = *(const float2*)&ldsX[SWZ(r, kbase)];
        v2f a; a.x = ap.x; a.y = ap.y;
        // B pair from global (L2-resident after first row-tile pass).
        v2f b;
        b.x = T[(kbase + 0) * COLS + colBase + r];
        b.y = T[(kbase + 1) * COLS + colBase + r];
        if (k0 + 8 < IN_F)
            __builtin_prefetch(&T[(kbase + 8) * COLS + colBase + r], 0, 1);

        acc = __builtin_amdgcn_wmma_f32_16x16x4_f32(
            /*neg_a=*/false, a, /*neg_b=*/false, b,
            /*c_mod=*/(short)0, acc, /*reuse_a=*/false, /*reuse_b=*/false);
    }

    // D layout: VGPR i -> (M=i, N=lane) for lanes 0-15, (M=i+8, N=lane-16).
    const int nOff = (lane < 16) ? 0 : 8;
    #pragma unroll
    for (int i = 0; i < 8; ++i)
        M[(rowBase + nOff + i) * COLS + colBase + r] = acc[i];
}

// ---------------------------------------------------------------------------
// Phase 2: for each o: o_b[m,o] = sum_n exp(-sum_k |M[n,o,k]-M[m,o,k]|) - 1
// One block per o (128 blocks), thread m owns row m; M[:,o,:] lives in LDS.
// ---------------------------------------------------------------------------
__global__ __launch_bounds__(256) void mbd_pair(const float* __restrict__ M,
                                                float* __restrict__ out)
{
    __shared__ float rows[N_SAMP * (KDIM + 1)];   // pad -> conflict-free own-row read
    const int o = blockIdx.x;
    const int m = threadIdx.x;

    for (int idx = m; idx < N_SAMP * KDIM; idx += 256) {
        const int n = idx >> 5;
        const int k = idx & (KDIM - 1);
        rows[n * (KDIM + 1) + k] = M[n * COLS + o * KDIM + k];
    }
    __syncthreads();

    float my[KDIM];
    #pragma unroll
    for (int k = 0; k < KDIM; ++k) my[k] = rows[m * (KDIM + 1) + k];

    float acc = 0.f;
    for (int n = 0; n < N_SAMP; ++n) {
        const float* rn = &rows[n * (KDIM + 1)];   // broadcast reads
        float norm = 0.f;
        #pragma unroll
        for (int k = 0; k < KDIM; ++k)
            norm += fabsf(rn[k] - my[k]);
        acc += __expf(-norm);
    }
    out[m * OUT_STRIDE + IN_F + o] = acc - 1.0f;   // subtract self-term exp(0)
}

// ---------------------------------------------------------------------------
// Phase 3: out[n, 0:1024] = x
// ---------------------------------------------------------------------------
__global__ void mbd_copyx(const float* __restrict__ x, float* __restrict__ out)
{
    const int idx = blockIdx.x * blockDim.x + threadIdx.x;
    if (idx < N_SAMP * IN_F) {
        const int n = idx >> 10;
        const int c = idx & (IN_F - 1);
        out[n * OUT_STRIDE + c] = x[idx];
    }
}

extern "C" void kernel_launch(void* const* d_in, const int* in_sizes, int n_in,
                              void* d_out, int out_size, void* d_ws, size_t ws_size,
                              hipStream_t stream) {
    (void)in_sizes; (void)n_in; (void)out_size; (void)ws_size;
    const float* x = (const float*)d_in[0];
    const float* T = (const float*)d_in[1];
    float* out = (float*)d_out;
    float* M   = (float*)d_ws;               // 256*4096*4 = 4 MB scratch

    mbd_gemm<<<dim3(64, 16), 128, 0, stream>>>(x, T, M);
    mbd_copyx<<<(N_SAMP * IN_F + 255) / 256, 256, 0, stream>>>(x, out);
    mbd_pair<<<OUT_F, 256, 0, stream>>>(M, out);
}